// Operator2DBase_Gradient_Kxy_15324443312426
// MI455X (gfx1250) — compile-verified
//
#include <hip/hip_runtime.h>

// CDNA5 / gfx1250: fused spatially-varying 5x5 stencil + central-difference
// update, with the banded per-row contraction mapped onto fp32 WMMA
// (V_WMMA_F32_16X16X4_F32) so the matrix pipe does the 25-tap sums at full
// fp32 precision.  Memory-bound problem (~21MB in / 8MB out, fits in L2).
//
// Round 2 change: masked operand loads are now BRANCHLESS (clamp the index so
// the address is always valid, load unconditionally, v_cndmask the result).
// Round 1 emitted an exec-mask branch + s_wait_xcnt around every single
// global_load_b32, which dominated the instruction mix (salu=2141, wait=631).

typedef float v2f __attribute__((ext_vector_type(2)));
typedef float v8f __attribute__((ext_vector_type(8)));

#define Hdim 256
#define Wdim 256
#define HWsz (Hdim * Wdim)
#define RROWS 8

// Branchless masked load of kernel weight K[di, dj, h, x]:
// clamp indices to a valid address, unconditional load, select 0 if OOB.
__device__ __forceinline__ float kload(const float* __restrict__ Kp,
                                       int di, int dj, int hrow, int x) {
    const bool ok = ((unsigned)dj <= 4u) & ((unsigned)x < (unsigned)Wdim);
    const int djc = ok ? dj : 0;                 // v_cndmask, keeps addr valid
    const int xc  = ok ? x  : 0;
    const float v = Kp[(di * 5 + djc) * HWsz + hrow * Wdim + xc];
    return ok ? v : 0.0f;
}

// Branchless zero-padded load of f[b, srcrow, y].
__device__ __forceinline__ float floadz(const float* __restrict__ fp,
                                        int bidx, int srcrow, int y) {
    const bool ok = ((unsigned)y < (unsigned)Wdim);
    const int yc  = ok ? y : 0;
    const float v = fp[bidx * HWsz + srcrow * Wdim + yc];
    return ok ? v : 0.0f;
}

// Compute one 16-column tile of fk(row hrow, cols [Xp, Xp+16)) for a 16-batch
// half, as D[m = x_local, n = b_local] via chained V_WMMA_F32_16X16X4_F32.
//   A[m,k] = K[di, y-x+2, hrow, x]   (banded weights; shared across batches)
//   B[k,n] = f[b, hrow+di-2, y]      (y = ybase + k)
// Per-lane operand placement matches the ISA 16x16x4 f32 layouts:
//   A: lane L -> m = L&15, k = 2*(L>>4)+{0,1}
//   B: lane L -> n = L&15, k = 2*(L>>4)+{0,1}
//   D: lane L, component r -> m = r + 8*(L>>4), n = L&15
__device__ __forceinline__ v8f fk_tile(const float* __restrict__ Kp,
                                       const float* __restrict__ fp,
                                       int hrow, int Xp, int m, int half,
                                       int bidx) {
    v8f acc = {0.f, 0.f, 0.f, 0.f, 0.f, 0.f, 0.f, 0.f};
    if (hrow < 0 || hrow >= Hdim) return acc;   // zero-pad of fkx/fky in h (uniform)
    const int xA = Xp + m;
#pragma unroll
    for (int di = 0; di < 5; ++di) {
        const int srcrow = hrow + di - 2;       // uniform across the wave
        if (srcrow < 0 || srcrow >= Hdim) continue;  // zero-pad of f in h (uniform)
#pragma unroll
        for (int c = 0; c < 5; ++c) {
            const int ybase = Xp - 2 + 4 * c;
            const int y0 = ybase + 2 * half;
            v2f A, Bv;
            A.x  = kload(Kp, di, y0 - xA + 2,     hrow, xA);
            A.y  = kload(Kp, di, y0 + 1 - xA + 2, hrow, xA);
            Bv.x = floadz(fp, bidx, srcrow, y0);
            Bv.y = floadz(fp, bidx, srcrow, y0 + 1);
            acc = __builtin_amdgcn_wmma_f32_16x16x4_f32(
                false, A, false, Bv, (short)0, acc, false, false);
        }
    }
    return acc;
}

extern "C" __global__ __launch_bounds__(64)
void grad_kxy_wmma_kernel(const float* __restrict__ f,
                          const float* __restrict__ dt,
                          const float* __restrict__ Kx,
                          const float* __restrict__ Ky,
                          float* __restrict__ out) {
    const int lane = threadIdx.x & 31;
    const int g    = threadIdx.x >> 5;   // batch half: wave 0 -> b 0..15, wave 1 -> b 16..31
    const int m    = lane & 15;
    const int half = lane >> 4;
    const int X    = blockIdx.x * 16;
    const int h0   = blockIdx.y * RROWS;
    const int bidx = g * 16 + m;
    const float dtb = dt[bidx];

    // Register-carried fkx rows (h-1, h, h+1) so each gx tile is computed once.
    v8f gx_m1 = fk_tile(Kx, f, h0 - 1, X, m, half, bidx);
    v8f gx_0  = fk_tile(Kx, f, h0,     X, m, half, bidx);

    for (int r = 0; r < RROWS; ++r) {
        const int h = h0 + r;

        // Prefetch next row's kernel data into cache (global_prefetch_b8).
        {
            const int hp = (h + 2 < Hdim) ? h + 2 : h;
            __builtin_prefetch(Kx + (size_t)hp * Wdim + X + lane, 0, 1);
            __builtin_prefetch(Ky + (size_t)hp * Wdim + X + lane, 0, 1);
        }

        v8f gx_p1 = fk_tile(Kx, f, h + 1, X,     m, half, bidx);  // fkx row h+1, cols [X,X+16)
        v8f gyA   = fk_tile(Ky, f, h,     X - 1, m, half, bidx);  // fky row h,  cols [X-1,X+15)
        v8f gyB   = fk_tile(Ky, f, h,     X + 1, m, half, bidx);  // fky row h,  cols [X+1,X+17)

        // Component r of every accumulator maps to output column
        // x = X + 8*half + r, so gyB - gyA is exactly fky(x+1)-fky(x-1) and
        // gx_p1 - gx_m1 is fkx(h+1)-fkx(h-1): no cross-lane shifts needed.
        const int xbase = X + 8 * half;
        const float* frow = f   + (size_t)bidx * HWsz + (size_t)h * Wdim + xbase;
        float*       orow = out + (size_t)bidx * HWsz + (size_t)h * Wdim + xbase;
        const float4 fin0 = *(const float4*)(frow);
        const float4 fin1 = *(const float4*)(frow + 4);
        const float fin[8] = {fin0.x, fin0.y, fin0.z, fin0.w,
                              fin1.x, fin1.y, fin1.z, fin1.w};
        float vals[8];
#pragma unroll
        for (int rr = 0; rr < 8; ++rr) {
            const float df = 0.5f * ((gx_p1[rr] - gx_m1[rr]) +
                                     (gyB[rr]   - gyA[rr]));
            const float o = fin[rr] + df * dtb;
            vals[rr] = o > 0.0f ? o : 0.0f;
        }
        *(float4*)(orow)     = make_float4(vals[0], vals[1], vals[2], vals[3]);
        *(float4*)(orow + 4) = make_float4(vals[4], vals[5], vals[6], vals[7]);

        gx_m1 = gx_0;
        gx_0  = gx_p1;
    }
}

extern "C" void kernel_launch(void* const* d_in, const int* in_sizes, int n_in,
                              void* d_out, int out_size, void* d_ws, size_t ws_size,
                              hipStream_t stream) {
    (void)in_sizes; (void)n_in; (void)out_size; (void)d_ws; (void)ws_size;
    const float* f  = (const float*)d_in[0];
    const float* dt = (const float*)d_in[1];
    const float* Kx = (const float*)d_in[2];
    const float* Ky = (const float*)d_in[3];
    float* out = (float*)d_out;

    dim3 grid(Wdim / 16, Hdim / RROWS, 1);  // 16 x-tiles x 32 row strips
    dim3 block(64, 1, 1);                   // 2 waves: one per 16-batch half
    hipLaunchKernelGGL(grad_kxy_wmma_kernel, grid, block, 0, stream,
                       f, dt, Kx, Ky, out);
}